// HSTUPositionalEncoder_10514079941191
// MI455X (gfx1250) — compile-verified
//
#include <hip/hip_runtime.h>
#include <math.h>

// HSTU positional encoder for MI455X (gfx1250, wave32).
// out[i,:] = emb[i,:]*sqrt(D) + pos_w[pos_ind(i),:] + ts_w[ts_ind(i),:]
//
// One wave32 per token row (D=512 floats = 128 float4 -> 4 float4 per lane).
// Per-row index math is wave-uniform (readfirstlane -> SALU/SMEM).
// emb/out use non-temporal hints (one-shot 64 MB streams) so the 24 MB of
// gather tables stay resident in the 192 MB L2; table gathers stay RT.
// Memory-bound op: ~128 MB HBM-critical traffic -> ~5.5 us at 23.3 TB/s.
// No contraction dimension exists, so WMMA is not applicable here.

typedef float v4f __attribute__((ext_vector_type(4)));

#define ROWS_PER_BLOCK 8
#define THREADS (ROWS_PER_BLOCK * 32)
#define CHUNKS 4   // D4 == 128 float4 per row / 32 lanes

__global__ __launch_bounds__(THREADS) void hstu_pos_enc_kernel(
    const float* __restrict__ emb,        // [L, D]
    const float* __restrict__ pos_w,      // [P, D]
    const float* __restrict__ ts_w,       // [NTB+1, D]
    const int*   __restrict__ seq_offsets,// [B+1]
    const int*   __restrict__ seq_lengths,// [B]
    const int*   __restrict__ num_targets,// [B]
    const int*   __restrict__ seq_ts,     // [L]
    float*       __restrict__ out,        // [L, D]
    int L, int D4, int B, int P, int ntb_rows, float alpha)
{
    const int lane = threadIdx.x & 31;
    // Force the row id into an SGPR: all row-derived (lane-independent) math
    // becomes SALU, and metadata lookups become SMEM s_loads.
    const int row = __builtin_amdgcn_readfirstlane(
        blockIdx.x * ROWS_PER_BLOCK + (threadIdx.x >> 5));
    if (row >= L) return;

    // ---- uniform per-row index math (pure scalar path) ----
    // Branchless batch search: fixed trip count -> s_load/s_cmp/s_add chain,
    // no exec-mask loop machinery.
    int b = 0;
    for (int j = 1; j < B; ++j)
        b += (seq_offsets[j] <= row) ? 1 : 0;

    const int off_b  = seq_offsets[b];
    const int off_b1 = seq_offsets[b + 1];
    const int rel    = row - off_b;

    int high = seq_lengths[b]; if (high < 0) high = 0;
    int nt   = num_targets[b]; if (nt   < 0) nt   = 0;
    high -= nt;                if (high < 0) high = 0;

    // MAX_CONTEXTUAL_SEQ_LEN == 0 -> the rel<0 branch in the reference is dead.
    int pos = (rel < high) ? rel : high;
    pos = high - pos;
    if (pos > P - 1) pos = P - 1;
    if (pos < 0)     pos = 0;

    // time-bucket index: trunc(sqrt(max(qt - tt, 1e-6) / 60))
    const float qt = (float)seq_ts[off_b1 - 1];
    const float tt = (float)seq_ts[row];
    float dt = fmaxf(qt - tt, 1e-6f) / 60.0f;
    dt = sqrtf(dt);
    int tsi = (int)dt;                      // dt >= 0, trunc == astype(int32)
    if (tsi < 0)            tsi = 0;
    if (tsi > ntb_rows - 1) tsi = ntb_rows - 1;

    // ---- vector streaming: grouped 128-bit loads, one wait, then stores ----
    const v4f* __restrict__ e = (const v4f*)emb   + (size_t)row * D4;
    const v4f* __restrict__ p = (const v4f*)pos_w + (size_t)pos * D4;
    const v4f* __restrict__ t = (const v4f*)ts_w  + (size_t)tsi * D4;
    v4f*       __restrict__ o = (v4f*)out         + (size_t)row * D4;

    // Lane-strided prefetch covering the full 2 KB of each gather row
    // (gfx1250 global_prefetch_b8): hides index-math -> gather latency.
    __builtin_prefetch((const char*)p + (size_t)lane * 64, 0, 3);
    __builtin_prefetch((const char*)t + (size_t)lane * 64, 0, 3);

    v4f ev[CHUNKS], pv[CHUNKS], tv[CHUNKS];

    // Issue all 12 b128 loads back-to-back: maximal MLP, single wait point.
#pragma unroll
    for (int k = 0; k < CHUNKS; ++k) {
        const int c = k * 32 + lane;
        pv[k] = p[c];                             // RT: L2-resident table
        tv[k] = t[c];                             // RT: L2-resident table
    }
#pragma unroll
    for (int k = 0; k < CHUNKS; ++k) {
        const int c = k * 32 + lane;
        ev[k] = __builtin_nontemporal_load(&e[c]); // NT: one-shot stream
    }

#pragma unroll
    for (int k = 0; k < CHUNKS; ++k) {
#pragma clang fp contract(off)
        const int c = k * 32 + lane;
        v4f sum = pv[k] + tv[k];         // matches reference: x + (pos + ts)
        v4f r   = ev[k] * alpha + sum;   // contraction off -> reference rounding
        __builtin_nontemporal_store(r, &o[c]);   // NT: keep tables in L2
    }
}

extern "C" void kernel_launch(void* const* d_in, const int* in_sizes, int n_in,
                              void* d_out, int out_size, void* d_ws, size_t ws_size,
                              hipStream_t stream) {
    // setup_inputs() order:
    // 0: max_seq_len (int,1)   1: seq_lengths (int,B)   2: seq_offsets (int,B+1)
    // 3: seq_embeddings (f32,L*D)  4: num_targets (int,B)  5: seq_timestamps (int,L)
    // 6: pos_weight (f32,P*D)  7: ts_weight (f32,(NTB+1)*D)
    const int*   seq_lengths = (const int*)d_in[1];
    const int*   seq_offsets = (const int*)d_in[2];
    const float* emb         = (const float*)d_in[3];
    const int*   num_targets = (const int*)d_in[4];
    const int*   seq_ts      = (const int*)d_in[5];
    const float* pos_w       = (const float*)d_in[6];
    const float* ts_w        = (const float*)d_in[7];
    float*       out         = (float*)d_out;

    const int B = in_sizes[1];
    const int L = in_sizes[5];
    const int D = (L > 0) ? in_sizes[3] / L : 0;
    const int D4 = D / 4;
    const int P = (D > 0) ? in_sizes[6] / D : 1;
    const int ntb_rows = (D > 0) ? in_sizes[7] / D : 1;  // NTB + 1
    const float alpha = sqrtf((float)D);

    const int blocks = (L + ROWS_PER_BLOCK - 1) / ROWS_PER_BLOCK;
    if (blocks <= 0) return;

    hstu_pos_enc_kernel<<<blocks, THREADS, 0, stream>>>(
        emb, pos_w, ts_w, seq_offsets, seq_lengths, num_targets, seq_ts,
        out, L, D4, B, P, ntb_rows, alpha);
}